// FeatureSimilarity_24799141167804
// MI455X (gfx1250) — compile-verified
//
#include <hip/hip_runtime.h>
#include <hip/hip_bf16.h>
#include <math.h>

typedef float v2f __attribute__((ext_vector_type(2)));
typedef float v4f __attribute__((ext_vector_type(4)));
typedef float v8f __attribute__((ext_vector_type(8)));

#define FEAT_D     128
#define TILE       128
#define LDS_STRIDE 132   // 128 + 4 pad: 132 % 64 == 4 -> conflict-free ds_load_b64, keeps 16B align

// ---------------- Kernel 1: per-row squared norms -> d_ws ----------------
__global__ __launch_bounds__(256)
void FeatureSimilarity_sqnorm_kernel(const float* __restrict__ f,
                                     float* __restrict__ sq, int n) {
    int gid  = blockIdx.x * blockDim.x + threadIdx.x;
    int row  = gid >> 5;               // one wave32 per row
    int lane = threadIdx.x & 31;
    if (row >= n) return;
    v4f v = *(const v4f*)(f + (size_t)row * FEAT_D + lane * 4);
    float s = v.x * v.x + v.y * v.y + v.z * v.z + v.w * v.w;
    s += __shfl_xor(s, 16, 32);
    s += __shfl_xor(s,  8, 32);
    s += __shfl_xor(s,  4, 32);
    s += __shfl_xor(s,  2, 32);
    s += __shfl_xor(s,  1, 32);
    if (lane == 0) sq[row] = s;
}

// ---------------- Kernel 2: WMMA gram + fused -sqrt epilogue ----------------
__global__ __launch_bounds__(256)
void FeatureSimilarity_pairdist_kernel(const float* __restrict__ f,
                                       const float* __restrict__ sq,
                                       float* __restrict__ out, int n) {
    __shared__ float ldsA[TILE * LDS_STRIDE];   // A tile, pre-scaled by -2
    __shared__ float ldsB[TILE * LDS_STRIDE];   // B tile (features rows j)

    const int tileI = blockIdx.y * TILE;   // output rows
    const int tileJ = blockIdx.x * TILE;   // output cols
    const int t = threadIdx.x;

    // Stage both 128x128 fp32 tiles: 16 float4s per thread per tile.
    #pragma unroll
    for (int c = 0; c < 16; ++c) {
        int flat = (c * 256 + t) * 4;
        int row  = flat >> 7;      // /128
        int k    = flat & 127;
        v4f a = *(const v4f*)(f + (size_t)(tileI + row) * FEAT_D + k);
        v4f b = *(const v4f*)(f + (size_t)(tileJ + row) * FEAT_D + k);
        a.x *= -2.f; a.y *= -2.f; a.z *= -2.f; a.w *= -2.f;   // fold the -2*gram
        *(v4f*)(ldsA + row * LDS_STRIDE + k) = a;
        *(v4f*)(ldsB + row * LDS_STRIDE + k) = b;
    }
    __syncthreads();

    const int lane    = t & 31;
    const int wave    = t >> 5;
    const int waveRow = (wave >> 2) * 64;   // 2 wave-rows of 64
    const int waveCol = (wave & 3) * 32;    // 4 wave-cols of 32
    const int laneLow = lane & 15;
    const int halfSel = lane >> 4;          // which K pair / which M half

    v8f acc[4][2];
    #pragma unroll
    for (int r = 0; r < 4; ++r)
        #pragma unroll
        for (int c2 = 0; c2 < 2; ++c2) {
            v8f z = {};
            acc[r][c2] = z;
        }

    // A fragment (16x4 f32): lane holds M = lane&15, K = kstep*4 + (lane>>4)*2 + {0,1}
    // B fragment (4x16 f32): lane holds N = lane&15, same K pair -> identical addressing
    const float* aBase = ldsA + (waveRow + laneLow) * LDS_STRIDE + halfSel * 2;
    const float* bBase = ldsB + (waveCol + laneLow) * LDS_STRIDE + halfSel * 2;

    #pragma unroll
    for (int k = 0; k < FEAT_D / 4; ++k) {
        v2f afrag[4], bfrag[2];
        #pragma unroll
        for (int r = 0; r < 4; ++r)
            afrag[r] = *(const v2f*)(aBase + r * 16 * LDS_STRIDE + k * 4);
        #pragma unroll
        for (int c2 = 0; c2 < 2; ++c2)
            bfrag[c2] = *(const v2f*)(bBase + c2 * 16 * LDS_STRIDE + k * 4);
        #pragma unroll
        for (int r = 0; r < 4; ++r)
            #pragma unroll
            for (int c2 = 0; c2 < 2; ++c2)
                acc[r][c2] = __builtin_amdgcn_wmma_f32_16x16x4_f32(
                    /*neg_a=*/false, afrag[r],
                    /*neg_b=*/false, bfrag[c2],
                    /*c_mod=*/(short)0, acc[r][c2],
                    /*reuse_a=*/false, /*reuse_b=*/false);
    }

    // Epilogue: d2 = sqA + sqB + (-2*gram); out = -sqrt(max(d2,0))
    // C/D layout: VGPR v -> M = v + 8*(lane>>4); N = lane&15
    #pragma unroll
    for (int r = 0; r < 4; ++r) {
        const int mBase = tileI + waveRow + r * 16 + halfSel * 8;
        float sqA[8];
        #pragma unroll
        for (int v = 0; v < 8; ++v) sqA[v] = sq[mBase + v];
        #pragma unroll
        for (int c2 = 0; c2 < 2; ++c2) {
            const int nCol = tileJ + waveCol + c2 * 16 + laneLow;
            const float sqB = sq[nCol];
            #pragma unroll
            for (int v = 0; v < 8; ++v) {
                float d2 = sqA[v] + sqB + acc[r][c2][v];
                d2 = fmaxf(d2, 0.0f);
                out[(size_t)(mBase + v) * n + nCol] = -sqrtf(d2);
            }
        }
    }
}

extern "C" void kernel_launch(void* const* d_in, const int* in_sizes, int n_in,
                              void* d_out, int out_size, void* d_ws, size_t ws_size,
                              hipStream_t stream) {
    const float* f  = (const float*)d_in[0];
    float* out      = (float*)d_out;
    float* sq       = (float*)d_ws;           // n floats of scratch
    const int n     = in_sizes[0] / FEAT_D;   // 8192

    // Kernel 1: one wave32 per row
    int threads = 256;
    int blocks  = (n * 32 + threads - 1) / threads;
    FeatureSimilarity_sqnorm_kernel<<<blocks, threads, 0, stream>>>(f, sq, n);

    // Kernel 2: 128x128 output tile per block
    dim3 grid(n / TILE, n / TILE);
    FeatureSimilarity_pairdist_kernel<<<grid, 256, 0, stream>>>(f, sq, out, n);
}